// ConvPool_6193342841562
// MI455X (gfx1250) — compile-verified
//
#include <hip/hip_runtime.h>
#include <cstdint>
#include <cstddef>

// Problem constants (fixed by the reference setup)
#define B_    4
#define N_    32768
#define P_    8192
#define S_    32
#define C_    64
#define CIN_  67          // 3 rel-pos + 64 features
#define COUT_ 128
#define KCH_  17          // K padded to 68 = 17 chunks of 4
#define NTOT_ 1048576     // B*P*S samples per channel for batch stats
#define BN_EPS_ 1e-5f

typedef float v2f __attribute__((ext_vector_type(2)));
typedef float v8f __attribute__((ext_vector_type(8)));

// ---------------------------------------------------------------- zero accum
__global__ void k_zero(float* __restrict__ acc) {
    acc[threadIdx.x] = 0.0f;   // <<<1,256>>> zeroes gAcc[0..255]
}

// ------------------------------------------------- transpose [B,C,N]->[B,N,C]
__global__ __launch_bounds__(256) void k_transpose(const float* __restrict__ feat,
                                                   float* __restrict__ featT) {
    __shared__ float tile[64][65];
    const int b  = blockIdx.x >> 9;            // N/64 = 512 tiles per batch
    const int n0 = (blockIdx.x & 511) << 6;
    const int t  = threadIdx.x;
    const int nn = t & 63, c0 = t >> 6;        // 4 c-rows per pass
    const size_t baseIn = (size_t)b * C_ * N_;
#pragma unroll
    for (int i = 0; i < 16; ++i) {
        const int c = c0 + (i << 2);
        tile[c][nn] = feat[baseIn + (size_t)c * N_ + n0 + nn];   // coalesced in n
    }
    __syncthreads();
    const int cc = t & 63, r0 = t >> 6;
    const size_t baseOut = ((size_t)b * N_ + n0) * C_;
#pragma unroll
    for (int i = 0; i < 16; ++i) {
        const int r = r0 + (i << 2);
        featT[baseOut + (size_t)r * C_ + cc] = tile[cc][r];      // coalesced in c
    }
}

// -------------------------------------------------------------- main fused op
// Per workgroup: 2 query points (64 samples) x all 128 output channels.
// WMMA f32 16x16x4 over K=68 (padded). Produces pre-BN max/min per (b,o,p)
// and accumulates per-channel sum / sum-of-squares for batch statistics.
__global__ __launch_bounds__(256) void k_main(const float* __restrict__ qxyz,
                                              const float* __restrict__ sxyz,
                                              const int*   __restrict__ nidx,
                                              const float* __restrict__ w,
                                              const float* __restrict__ featT,
                                              float* __restrict__ dMax,     // = d_out (pre-BN max)
                                              float* __restrict__ minBuf,   // pre-BN min
                                              float* __restrict__ gAcc) {   // [0:128]=sum [128:256]=sumsq
    __shared__ float agg[64][68];                 // 64 samples x (67+1 pad) channels
    __shared__ float chSum[COUT_], chSS[COUT_];

    const int tid = threadIdx.x;
    const int b   = blockIdx.x / (P_ / 2);
    const int p0  = (blockIdx.x % (P_ / 2)) * 2;

    if (tid < COUT_) { chSum[tid] = 0.0f; chSS[tid] = 0.0f; }

    // ---- Stage A: build agg tile in LDS (4 threads per sample) ----
    {
        const int m = tid >> 2, q = tid & 3;            // sample, quarter
        const int p = p0 + (m >> 5), s = m & 31;
        const int idx = nidx[((size_t)b * P_ + p) * S_ + s];
        const float4* fsrc = (const float4*)(featT + ((size_t)b * N_ + idx) * C_ + q * 16);
        const float4 v0 = fsrc[0], v1 = fsrc[1], v2 = fsrc[2], v3 = fsrc[3];
        float* dst = &agg[m][3 + q * 16];
        dst[0]=v0.x; dst[1]=v0.y; dst[2]=v0.z;  dst[3]=v0.w;
        dst[4]=v1.x; dst[5]=v1.y; dst[6]=v1.z;  dst[7]=v1.w;
        dst[8]=v2.x; dst[9]=v2.y; dst[10]=v2.z; dst[11]=v2.w;
        dst[12]=v3.x; dst[13]=v3.y; dst[14]=v3.z; dst[15]=v3.w;
        if (q == 0) {
            const float* sp = sxyz + ((size_t)b * N_ + idx) * 3;
            const float* qp = qxyz + ((size_t)b * P_ + p) * 3;
            agg[m][0] = sp[0] - qp[0];
            agg[m][1] = sp[1] - qp[1];
            agg[m][2] = sp[2] - qp[2];
            agg[m][67] = 0.0f;                          // K padding channel
        }
    }

    // ---- Stage B: per-wave weight B-fragments (reused for 4 M-subtiles) ----
    const int wave = tid >> 5, lane = tid & 31;
    const int n0 = wave * 16, nn = lane & 15, kh = lane >> 4;
    v2f bfrag[KCH_];
#pragma unroll
    for (int kk = 0; kk < KCH_; ++kk) {
        const int k0 = kk * 4 + kh * 2;
        bfrag[kk].x = (k0     < CIN_) ? w[(n0 + nn) * CIN_ + k0    ] : 0.0f;
        bfrag[kk].y = (k0 + 1 < CIN_) ? w[(n0 + nn) * CIN_ + k0 + 1] : 0.0f;
    }
    __syncthreads();

    // ---- Stage C: WMMA over K, fused stats + max/min reduction ----
    float pmax = -3.402823466e38f, pmin = 3.402823466e38f;
#pragma unroll
    for (int sub = 0; sub < 4; ++sub) {                 // subtiles 0,1 -> p0 ; 2,3 -> p0+1
        const int m0 = sub * 16;
        v8f acc = {};
#pragma unroll
        for (int kk = 0; kk < KCH_; ++kk) {
            v2f a;
            const float* ap = &agg[m0 + nn][kk * 4 + kh * 2];
            a.x = ap[0]; a.y = ap[1];
            acc = __builtin_amdgcn_wmma_f32_16x16x4_f32(
                false, a, false, bfrag[kk], (short)0, acc, false, false);
        }
        // per-lane reduce over 8 rows (channel = n0+nn, rows M = r + 8*kh)
        float ls = 0.0f, lss = 0.0f;
        float lmax = -3.402823466e38f, lmin = 3.402823466e38f;
#pragma unroll
        for (int r = 0; r < 8; ++r) {
            const float y = acc[r];
            ls += y; lss += y * y;
            lmax = fmaxf(lmax, y); lmin = fminf(lmin, y);
        }
        atomicAdd(&chSum[n0 + nn], ls);                 // ds_add_f32
        atomicAdd(&chSS[n0 + nn], lss);
        lmax = fmaxf(lmax, __shfl_xor(lmax, 16, 32));   // join M halves
        lmin = fminf(lmin, __shfl_xor(lmin, 16, 32));
        pmax = fmaxf(pmax, lmax);
        pmin = fminf(pmin, lmin);
        if (sub & 1) {                                  // finished one point's 32 samples
            if (lane < 16) {
                const size_t o = ((size_t)b * COUT_ + n0 + lane) * P_ + (p0 + (sub >> 1));
                dMax[o] = pmax;
                minBuf[o] = pmin;
            }
            pmax = -3.402823466e38f; pmin = 3.402823466e38f;
        }
    }

    __syncthreads();
    if (tid < COUT_)            unsafeAtomicAdd(&gAcc[tid], chSum[tid]);
    else if (tid < 2 * COUT_)   unsafeAtomicAdd(&gAcc[tid], chSS[tid - COUT_]);
}

// -------------------------------------------------- finalize: BN affine + ReLU
__global__ __launch_bounds__(256) void k_final(float* __restrict__ out,
                                               const float* __restrict__ minBuf,
                                               const float* __restrict__ gAcc,
                                               const float* __restrict__ gamma,
                                               const float* __restrict__ beta) {
    const size_t i = (size_t)blockIdx.x * 256 + threadIdx.x;
    const int o = (int)((i / P_) % COUT_);
    const float invN  = 1.0f / (float)NTOT_;
    const float mean  = gAcc[o] * invN;
    const float var   = gAcc[COUT_ + o] * invN - mean * mean;
    const float scale = gamma[o] * rsqrtf(var + BN_EPS_);
    const float shift = beta[o] - mean * scale;
    // max_s relu(scale*y+shift) = relu(scale * (scale>=0 ? max_s y : min_s y) + shift)
    const float m = (scale >= 0.0f) ? out[i] : minBuf[i];
    out[i] = fmaxf(scale * m + shift, 0.0f);
}

// ---------------------------------------------------------------------- launch
extern "C" void kernel_launch(void* const* d_in, const int* in_sizes, int n_in,
                              void* d_out, int out_size, void* d_ws, size_t ws_size,
                              hipStream_t stream) {
    const float* qxyz  = (const float*)d_in[0];
    const float* sxyz  = (const float*)d_in[1];
    const float* feat  = (const float*)d_in[2];
    const int*   nidx  = (const int*)  d_in[3];
    const float* w     = (const float*)d_in[4];
    const float* gamma = (const float*)d_in[5];
    const float* beta  = (const float*)d_in[6];
    float* out = (float*)d_out;

    // workspace: featT [B,N,C] (32 MB) | minBuf [B,Cout,P] (16 MB) | gAcc[256]
    float* featT  = (float*)d_ws;
    float* minBuf = featT + (size_t)B_ * N_ * C_;
    float* gAcc   = minBuf + (size_t)B_ * COUT_ * P_;

    k_zero     <<<1,                    256, 0, stream>>>(gAcc);
    k_transpose<<<B_ * (N_ / 64),       256, 0, stream>>>(feat, featT);
    k_main     <<<B_ * (P_ / 2),        256, 0, stream>>>(qxyz, sxyz, nidx, w, featT,
                                                          out, minBuf, gAcc);
    k_final    <<<(B_ * COUT_ * P_)/256, 256, 0, stream>>>(out, minBuf, gAcc, gamma, beta);
}